// STDHMP_31782757991130
// MI455X (gfx1250) — compile-verified
//
#include <hip/hip_runtime.h>
#include <math.h>

#define BB 16
#define TT 16
#define KK 4
#define VV 10003
#define DD 256
#define CC 64
#define T2 14
#define NEG_INF (-3.4e38f)

#define NTILE 626          // ceil(10003/16)
#define CHUNKS 16          // V-dimension split across workgroups
#define TPC 40             // tiles per chunk (16*40 = 640 >= 626)

typedef __attribute__((ext_vector_type(2))) float v2f;
typedef __attribute__((ext_vector_type(8))) float v8f;

// ---------------------------------------------------------------- init chain
__global__ void k_init_chain(const int* __restrict__ loc_tar, int* __restrict__ chain) {
    int t = threadIdx.x;                 // 256 threads, 1 block
    if (t < BB * TT) chain[t] = 0;       // only first 16 rows are read at step 0
    __syncthreads();
    if (t < BB) chain[t * TT] = loc_tar[t * TT];
}

// ---------------------------------------------------------------- build X = E[lid] + zs[b]
__global__ void k_build_x(const int* __restrict__ chain, const float* __restrict__ zs,
                          const float* __restrict__ E, float* __restrict__ X,
                          int step, int M) {
    int n = blockIdx.x, d = threadIdx.x;        // grid=N, block=256
    int lid = chain[n * TT + step];
    int b = n / M;
    X[n * DD + d] = E[lid * DD + d] + zs[b * DD + d];
}

// ---------------------------------------------------------------- GEMM (f32 WMMA) + online top4 + logsumexp
// grid = (CHUNKS, N/16); each block handles a 16-row tile x one V-chunk,
// writes one merged partial state (10 words) per row to PS.
#define XST 260   // LDS row stride for X tile (bank-conflict free: 260 % 64 == 4)
#define LST 17    // LDS row stride for logits scratch

__global__ void k_gemm_topk_part(const float* __restrict__ X, const float* __restrict__ Wv,
                                 const float* __restrict__ bv, float* __restrict__ PS) {
    __shared__ float Xs[16 * XST];         // 16 rows x 256 K of X
    __shared__ float Ls[8 * 16 * LST];     // per-wave 16x16 logits scratch
    __shared__ float Red[8 * 32 * 10];     // per-(wave,lane) partial state

    const int tid  = threadIdx.x;          // 256 threads = 8 waves
    const int wave = tid >> 5;
    const int lane = tid & 31;
    const int ch   = blockIdx.x;           // V chunk
    const int row0 = blockIdx.y * 16;

    const int tile_beg = ch * TPC;
    const int tile_end = (tile_beg + TPC < NTILE) ? (tile_beg + TPC) : NTILE;

    // stage X tile into LDS
    for (int j = 0; j < 16; ++j) {
        int e = tid + j * 256;             // e in [0, 4096)
        Xs[(e >> 8) * XST + (e & 255)] = X[row0 * DD + e];
    }
    __syncthreads();

    const int m16  = lane & 15;
    const int hi   = lane >> 4;            // 0: K=0,1 / cols 0-7   1: K=2,3 / cols 8-15
    const int kofs = hi * 2;

    float t0 = NEG_INF, t1 = NEG_INF, t2 = NEG_INF, t3 = NEG_INF;
    int   i0 = 0x7fffffff, i1 = 0x7fffffff, i2 = 0x7fffffff, i3 = 0x7fffffff;
    float mx = NEG_INF, sm = 0.f;

    for (int ct = tile_beg + wave; ct < tile_end; ct += 8) {
        const int col0  = ct * 16;
        const int col   = col0 + m16;
        const bool colok = (col < VV);
        const int colc  = colok ? col : (VV - 1);

        v8f acc = {0.f, 0.f, 0.f, 0.f, 0.f, 0.f, 0.f, 0.f};
        for (int k0 = 0; k0 < DD; k0 += 4) {
            v2f a, b;
            a.x = Xs[m16 * XST + k0 + kofs];
            a.y = Xs[m16 * XST + k0 + kofs + 1];
            float w0 = Wv[(k0 + kofs) * VV + colc];
            float w1 = Wv[(k0 + kofs + 1) * VV + colc];
            b.x = colok ? w0 : 0.f;
            b.y = colok ? w1 : 0.f;
            acc = __builtin_amdgcn_wmma_f32_16x16x4_f32(false, a, false, b,
                                                        (short)0, acc, false, false);
        }
        float bias = colok ? bv[colc] : 0.f;

        // scatter this wave's 16x16 tile to LDS (C/D layout -> row major)
        #pragma unroll
        for (int vr = 0; vr < 8; ++vr) {
            int m = vr + hi * 8;
            Ls[(wave * 16 + m) * LST + m16] = acc[vr] + bias;
        }
        __asm__ volatile("s_wait_dscnt 0x0" ::: "memory");

        // online update: lane owns row m16, columns [hi*8, hi*8+8)
        #pragma unroll
        for (int j = 0; j < 8; ++j) {
            int cl = hi * 8 + j;
            int c = col0 + cl;
            if (c < VV) {
                float v = Ls[(wave * 16 + m16) * LST + cl];
                if (v > mx) { sm = sm * expf(mx - v) + 1.f; mx = v; }
                else        { sm += expf(v - mx); }
                if (v > t3) {        // within-lane indices increase -> strict > is stable
                    if (v > t0)      { t3=t2;i3=i2; t2=t1;i2=i1; t1=t0;i1=i0; t0=v;i0=c; }
                    else if (v > t1) { t3=t2;i3=i2; t2=t1;i2=i1; t1=v;i1=c; }
                    else if (v > t2) { t3=t2;i3=i2; t2=v;i2=c; }
                    else             { t3=v;i3=c; }
                }
            }
        }
    }

    float* rp = &Red[(wave * 32 + lane) * 10];
    rp[0] = t0; rp[1] = t1; rp[2] = t2; rp[3] = t3;
    rp[4] = mx; rp[5] = sm;
    rp[6] = __int_as_float(i0); rp[7] = __int_as_float(i1);
    rp[8] = __int_as_float(i2); rp[9] = __int_as_float(i3);
    __syncthreads();

    if (tid < 16) {
        const int r = tid;
        float T0 = NEG_INF, T1 = NEG_INF, T2v = NEG_INF, T3 = NEG_INF;
        int   I0 = 0x7fffffff, I1 = 0x7fffffff, I2 = 0x7fffffff, I3 = 0x7fffffff;
        float MX = NEG_INF, SM = 0.f;
        for (int w = 0; w < 8; ++w)
            for (int h = 0; h < 2; ++h) {
                const float* sp = &Red[(w * 32 + h * 16 + r) * 10];
                float pmx = sp[4], psm = sp[5];
                if (pmx > NEG_INF) {
                    if (pmx > MX) { SM = SM * expf(MX - pmx) + psm; MX = pmx; }
                    else          { SM += psm * expf(pmx - MX); }
                }
                for (int e = 0; e < 4; ++e) {
                    float v = sp[e];
                    int   c = __float_as_int(sp[6 + e]);
                    if (v > NEG_INF) {   // stable insert: index tie-break (lower index wins)
                        bool b3 = (v > T3)  || (v == T3  && c < I3);
                        if (b3) {
                            bool b0 = (v > T0)  || (v == T0  && c < I0);
                            bool b1 = (v > T1)  || (v == T1  && c < I1);
                            bool b2 = (v > T2v) || (v == T2v && c < I2);
                            if (b0)      { T3=T2v;I3=I2; T2v=T1;I2=I1; T1=T0;I1=I0; T0=v;I0=c; }
                            else if (b1) { T3=T2v;I3=I2; T2v=T1;I2=I1; T1=v;I1=c; }
                            else if (b2) { T3=T2v;I3=I2; T2v=v;I2=c; }
                            else         { T3=v;I3=c; }
                        }
                    }
                }
            }
        // write merged per-chunk state for this row
        float* op = &PS[((size_t)(row0 + r) * CHUNKS + ch) * 10];
        op[0] = T0; op[1] = T1; op[2] = T2v; op[3] = T3;
        op[4] = MX; op[5] = SM;
        op[6] = __int_as_float(I0); op[7] = __int_as_float(I1);
        op[8] = __int_as_float(I2); op[9] = __int_as_float(I3);
    }
}

// ---------------------------------------------------------------- merge chunk states -> P (logp) and I (indices)
__global__ void k_reduce_topk(const float* __restrict__ PS, float* __restrict__ P,
                              int* __restrict__ I) {
    if (threadIdx.x != 0) return;
    int n = blockIdx.x;                    // grid = N rows
    float T0 = NEG_INF, T1 = NEG_INF, T2v = NEG_INF, T3 = NEG_INF;
    int   I0 = 0x7fffffff, I1 = 0x7fffffff, I2 = 0x7fffffff, I3 = 0x7fffffff;
    float MX = NEG_INF, SM = 0.f;
    for (int ch = 0; ch < CHUNKS; ++ch) {  // ascending chunk order = ascending columns
        const float* sp = &PS[((size_t)n * CHUNKS + ch) * 10];
        float pmx = sp[4], psm = sp[5];
        if (pmx > NEG_INF) {
            if (pmx > MX) { SM = SM * expf(MX - pmx) + psm; MX = pmx; }
            else          { SM += psm * expf(pmx - MX); }
        }
        for (int e = 0; e < 4; ++e) {
            float v = sp[e];
            int   c = __float_as_int(sp[6 + e]);
            if (v > NEG_INF) {
                bool b3 = (v > T3)  || (v == T3  && c < I3);
                if (b3) {
                    bool b0 = (v > T0)  || (v == T0  && c < I0);
                    bool b1 = (v > T1)  || (v == T1  && c < I1);
                    bool b2 = (v > T2v) || (v == T2v && c < I2);
                    if (b0)      { T3=T2v;I3=I2; T2v=T1;I2=I1; T1=T0;I1=I0; T0=v;I0=c; }
                    else if (b1) { T3=T2v;I3=I2; T2v=T1;I2=I1; T1=v;I1=c; }
                    else if (b2) { T3=T2v;I3=I2; T2v=v;I2=c; }
                    else         { T3=v;I3=c; }
                }
            }
        }
    }
    float lse = MX + logf(SM);
    P[n * 4 + 0] = T0 - lse; P[n * 4 + 1] = T1  - lse;
    P[n * 4 + 2] = T2v - lse; P[n * 4 + 3] = T3 - lse;
    I[n * 4 + 0] = I0; I[n * 4 + 1] = I1; I[n * 4 + 2] = I2; I[n * 4 + 3] = I3;
}

// ---------------------------------------------------------------- beam step 0
__global__ void k_beam0(const float* __restrict__ P, const int* __restrict__ I,
                        const int* __restrict__ loc_tar, int* __restrict__ cout,
                        float* __restrict__ pout) {
    if (threadIdx.x != 0) return;
    int b = blockIdx.x;
    for (int k = 0; k < KK; ++k) {
        int dst = b * KK + k;
        for (int t = 0; t < TT; ++t) cout[dst * TT + t] = 0;
        cout[dst * TT + 0] = loc_tar[b * TT];
        cout[dst * TT + 1] = I[b * KK + k];
        pout[dst] = P[b * KK + k];
    }
}

// ---------------------------------------------------------------- beam step i>=1 (stable top-8 of prob products)
__global__ void k_beam(const float* __restrict__ P, const int* __restrict__ I,
                       const int* __restrict__ cin, int* __restrict__ cout,
                       const float* __restrict__ pin, float* __restrict__ pout,
                       int step, int M) {
    if (threadIdx.x != 0) return;
    int b = blockIdx.x;
    int MK = M * KK;                       // <= 32
    float p2[32]; int used[32];
    for (int q = 0; q < MK; ++q) {
        int m = q / KK, k = q % KK;
        p2[q] = pin[b * M + m] * P[(b * M + m) * KK + k];   // product of log-probs (faithful)
        used[q] = 0;
    }
    for (int j = 0; j < 2 * KK; ++j) {
        int best = -1; float bvv = 0.f;
        for (int q = 0; q < MK; ++q)
            if (!used[q] && (best < 0 || p2[q] > bvv)) { best = q; bvv = p2[q]; }
        used[best] = 1;
        int m = best / KK, k = best % KK;
        int src = b * M + m, dst = b * 2 * KK + j;
        for (int t = 0; t < TT; ++t) cout[dst * TT + t] = cin[src * TT + t];
        cout[dst * TT + step + 1] = I[src * KK + k];
        pout[dst] = bvv;
    }
}

// ---------------------------------------------------------------- final: tanh(E@Wc+bc)@Wt + zt@Wzt
__global__ void k_final(const int* __restrict__ chain, const float* __restrict__ E,
                        const float* __restrict__ Wc, const float* __restrict__ bc,
                        const float* __restrict__ Wt, const float* __restrict__ zt,
                        const float* __restrict__ Wzt, float* __restrict__ out) {
    __shared__ float er[DD];
    __shared__ float red[CC];
    int b = blockIdx.x / T2, t = blockIdx.x % T2;   // grid = B*14, block = 64
    int c = threadIdx.x;
    int lid = chain[(b * 2 * KK + 0) * TT + (t + 1)];   // best beam, cols 1..T-2
    for (int j = 0; j < DD / CC; ++j) er[c + j * CC] = E[lid * DD + c + j * CC];
    __syncthreads();
    float s = bc[c];
    for (int d = 0; d < DD; ++d) s += er[d] * Wc[d * CC + c];
    float val = tanhf(s) * Wt[c];
    for (int j = 0; j < 4; ++j) { int d = c * 4 + j; val += zt[b * DD + d] * Wzt[d]; }
    red[c] = val;
    __syncthreads();
    for (int off = CC / 2; off > 0; off >>= 1) {
        if (c < off) red[c] += red[c + off];
        __syncthreads();
    }
    if (c == 0) {
        out[b * T2 + t] = (float)lid;                 // loc_chain (int values as float)
        out[BB * T2 + b * T2 + t] = red[0];           // tim_chain
    }
}

// ---------------------------------------------------------------- host
extern "C" void kernel_launch(void* const* d_in, const int* in_sizes, int n_in,
                              void* d_out, int out_size, void* d_ws, size_t ws_size,
                              hipStream_t stream) {
    (void)in_sizes; (void)n_in; (void)out_size; (void)ws_size;
    const int*   loc_tar = (const int*)  d_in[0];
    const float* zs      = (const float*)d_in[1];
    const float* zt      = (const float*)d_in[2];
    const float* E       = (const float*)d_in[3];
    const float* Wv      = (const float*)d_in[4];
    const float* bv      = (const float*)d_in[5];
    const float* Wc      = (const float*)d_in[6];
    const float* bc      = (const float*)d_in[7];
    const float* Wt      = (const float*)d_in[8];
    const float* Wzt     = (const float*)d_in[9];

    char* ws = (char*)d_ws;
    int*   chainA = (int*)  (ws + 0);        // 128*16 ints
    int*   chainB = (int*)  (ws + 8192);
    float* probA  = (float*)(ws + 16384);    // 128 floats
    float* probB  = (float*)(ws + 16896);
    float* P      = (float*)(ws + 17408);    // 128*4 floats
    int*   I      = (int*)  (ws + 19456);    // 128*4 ints
    float* X      = (float*)(ws + 21504);    // 128*256 floats
    float* PS     = (float*)(ws + 152576);   // 128*16*10 floats (chunk partial states)

    k_init_chain<<<1, 256, 0, stream>>>(loc_tar, chainA);
    for (int i = 0; i < TT - 1; ++i) {
        int N = (i == 0) ? BB : (i == 1 ? BB * KK : BB * 2 * KK);
        int M = N / BB;
        int*   cin  = (i & 1) ? chainB : chainA;
        int*   cout = (i & 1) ? chainA : chainB;
        float* pin  = (i & 1) ? probB  : probA;
        float* pout = (i & 1) ? probA  : probB;
        k_build_x<<<N, DD, 0, stream>>>(cin, zs, E, X, i, M);
        dim3 grid(CHUNKS, N / 16);
        k_gemm_topk_part<<<grid, 256, 0, stream>>>(X, Wv, bv, PS);
        k_reduce_topk<<<N, 32, 0, stream>>>(PS, P, I);
        if (i == 0) k_beam0<<<BB, 32, 0, stream>>>(P, I, loc_tar, cout, pout);
        else        k_beam <<<BB, 32, 0, stream>>>(P, I, cin, cout, pin, pout, i, M);
    }
    // last step i=14 is even -> wrote chainB
    k_final<<<BB * T2, CC, 0, stream>>>(chainB, E, Wc, bc, Wt, zt, Wzt, (float*)d_out);
}